// _3DFLCriterion_New_49667001811716
// MI455X (gfx1250) — compile-verified
//
#include <hip/hip_runtime.h>

// ---------------------------------------------------------------------------
// Problem constants (match the reference)
// ---------------------------------------------------------------------------
#define B_SZ 256
#define N_SZ 16384
#define COEF_LR 10.0f
#define START_LR 10

// Tiling: 1024 blocks = B * 4; each block owns 4096 points of one batch,
// processed as 4 tiles of 1024 points (256 threads * 4 points each).
#define BLOCKS_PER_BATCH 4
#define POINTS_PER_BLOCK (N_SZ / BLOCKS_PER_BATCH) // 4096
#define TILE_PTS 1024
#define NTILES (POINTS_PER_BLOCK / TILE_PTS)       // 4
#define NSTREAM 6                                  // c0.x/y/z (TDM) + p3.x/y/z (async)
#define SLOT_BYTES (TILE_PTS * 4)                  // 4096 B = one row of a tile
#define REGION_BYTES (NSTREAM * SLOT_BYTES)        // 24576 B per tile buffer
#define SMEM_BYTES (2 * REGION_BYTES)              // 49152 B double-buffered

typedef unsigned int v4u __attribute__((ext_vector_type(4)));
typedef unsigned int v8u __attribute__((ext_vector_type(8)));

// CDNA5 per-lane async DMA: global -> LDS, tracked by ASYNCcnt (per-wave, in-order).
__device__ __forceinline__ void async_ld_b128(unsigned lds_off, const float* gp) {
  unsigned long long ga = (unsigned long long)(uintptr_t)gp;
  asm volatile("global_load_async_to_lds_b128 %0, %1, off"
               :: "v"(lds_off), "v"(ga) : "memory");
}

// CDNA5 Tensor Data Mover: one descriptor DMAs a rows x TILE_PTS f32 tile
// (row stride = N_SZ elements) from global into LDS, tracked by TENSORcnt.
// 2D tensor -> only D# groups 0 and 1 are needed (VADDR2/3 = NULL form).
__device__ __forceinline__ void tdm_load_tile(unsigned lds_addr, const float* gp,
                                              unsigned rows) {
  unsigned long long ga = (unsigned long long)(uintptr_t)gp;
  v4u g0;
  g0.x = 1u;                                   // count=1, is_restore=0, gather=0
  g0.y = lds_addr;                             // lds_addr[31:0]
  g0.z = (unsigned)ga;                         // global_addr[31:0]
  g0.w = (unsigned)((ga >> 32) & 0x01FFFFFFull) | (2u << 30); // addr[56:32] | type=2
  v8u g1;
  g1.s0 = 2u << 16;                            // data_size=2 (4 bytes); no pad/iter/mask
  g1.s1 = (N_SZ & 0xFFFFu) << 16;              // tensor_dim0[15:0] in bits [63:48]
  g1.s2 = (N_SZ >> 16) | (rows << 16);         // tensor_dim0[31:16] | tensor_dim1[15:0]
  g1.s3 = TILE_PTS << 16;                      // tensor_dim1[31:16]=0 | tile_dim0
  g1.s4 = rows;                                // tile_dim1 | tile_dim2=0
  g1.s5 = N_SZ;                                // tensor_dim0_stride[31:0]
  g1.s6 = 0u;                                  // stride0[47:32] | stride1[15:0]
  g1.s7 = 0u;                                  // stride1[47:16]
  asm volatile("tensor_load_to_lds %0, %1" :: "s"(g0), "s"(g1) : "memory");
}

__global__ __launch_bounds__(256) void fl_criterion_main(
    const float* __restrict__ coord_pd0, const float* __restrict__ coord_pd1,
    const float* __restrict__ p3Ds,      const float* __restrict__ p3D_ids,
    const float* __restrict__ xys,       const float* __restrict__ pose,
    const float* __restrict__ camera,    float* __restrict__ ws) {
  __shared__ __align__(16) unsigned char smem[SMEM_BYTES];
  __shared__ float red[3][8];

  const int tid = threadIdx.x;
  const int b   = blockIdx.x / BLOCKS_PER_BATCH;
  const int sub = blockIdx.x % BLOCKS_PER_BATCH;
  const int p0  = sub * POINTS_PER_BLOCK;

  // Uniform (SGPR) wave id so the TDM issue/wait is under a scalar branch:
  // TDM ignores EXEC, so it must be skipped via s_cbranch, not predication.
  const bool is_wave0 = ((__builtin_amdgcn_readfirstlane(tid) >> 5) == 0);

  const size_t base3 = (size_t)b * 3 * N_SZ;
  const size_t base1 = (size_t)b * N_SZ;
  const float* c0 = coord_pd0 + base3;
  const float* q3 = p3Ds + base3;

  // Low 32 bits of a flat LDS pointer == byte offset within the group's LDS.
  const unsigned smb = (unsigned)(size_t)(void*)smem;

  // Per-batch constants: quaternion -> rotation, translation, intrinsics.
  const float* pp = pose + b * 7;
  const float t0 = pp[0], t1 = pp[1], t2 = pp[2];
  const float qw = pp[3], qx = pp[4], qy = pp[5], qz = pp[6];
  const float r00 = 1.f - 2.f*(qy*qy + qz*qz);
  const float r01 = 2.f*(qx*qy - qw*qz);
  const float r02 = 2.f*(qz*qx + qw*qy);
  const float r10 = 2.f*(qx*qy + qw*qz);
  const float r11 = 1.f - 2.f*(qx*qx + qz*qz);
  const float r12 = 2.f*(qy*qz - qw*qx);
  const float r20 = 2.f*(qz*qx - qw*qy);
  const float r21 = 2.f*(qy*qz + qw*qx);
  const float r22 = 1.f - 2.f*(qx*qx + qy*qy);
  const float* cam = camera + b * 5;
  const float fl = cam[2], ppx = cam[3], ppy = cam[4];

  float s1 = 0.f, su = 0.f, s2 = 0.f;

  // ---- pipeline prologue: stage tiles 0 and 1 ----
  if (is_wave0) {
    tdm_load_tile(smb + 0u * REGION_BYTES, c0 + p0, 3u);
    tdm_load_tile(smb + 1u * REGION_BYTES, c0 + p0 + TILE_PTS, 3u);
  }
#pragma unroll
  for (int r = 0; r < 2; ++r) {
    const int p = p0 + r * TILE_PTS + tid * 4;
    const unsigned lb = smb + (unsigned)(r * REGION_BYTES) + 3u * SLOT_BYTES + tid * 16;
    async_ld_b128(lb + 0u * SLOT_BYTES, q3 + 0 * N_SZ + p);
    async_ld_b128(lb + 1u * SLOT_BYTES, q3 + 1 * N_SZ + p);
    async_ld_b128(lb + 2u * SLOT_BYTES, q3 + 2 * N_SZ + p);
  }

  for (int t = 0; t < NTILES; ++t) {
    const int p = p0 + t * TILE_PTS + tid * 4;

    // Direct coalesced b128 loads for the light streams (issued before the
    // counter waits so they overlap with the DMA completion).
    const float4 wv   = *(const float4*)(p3D_ids   + base1 + p);
    const float4 uv   = *(const float4*)(coord_pd1 + base1 + p);
    const float4 xy01 = *(const float4*)(xys + 2*(base1 + p));
    const float4 xy23 = *(const float4*)(xys + 2*(base1 + p) + 4);

    // Wait for tile t's DMAs (both engines complete in order per wave):
    // steady state leaves exactly one tile in flight on each counter.
    if (t < NTILES - 1) {
      if (is_wave0) asm volatile("s_wait_tensorcnt 0x1" ::: "memory");
      asm volatile("s_wait_asynccnt 0x3" ::: "memory");
    } else {
      if (is_wave0) asm volatile("s_wait_tensorcnt 0x0" ::: "memory");
      asm volatile("s_wait_asynccnt 0x0" ::: "memory");
    }
    // TDM data was fetched by wave 0 on behalf of the whole block.
    __syncthreads();

    const unsigned char* rb = smem + (t & 1) * REGION_BYTES + tid * 16;
    const float4 cx = *(const float4*)(rb + 0 * SLOT_BYTES);
    const float4 cy = *(const float4*)(rb + 1 * SLOT_BYTES);
    const float4 cz = *(const float4*)(rb + 2 * SLOT_BYTES);
    const float4 gx = *(const float4*)(rb + 3 * SLOT_BYTES);
    const float4 gy = *(const float4*)(rb + 4 * SLOT_BYTES);
    const float4 gz = *(const float4*)(rb + 5 * SLOT_BYTES);

    const float* cxp = &cx.x; const float* cyp = &cy.x; const float* czp = &cz.x;
    const float* gxp = &gx.x; const float* gyp = &gy.x; const float* gzp = &gz.x;
    const float* wp  = &wv.x; const float* up  = &uv.x;
    const float xs[4] = {xy01.x, xy01.z, xy23.x, xy23.z};
    const float ys[4] = {xy01.y, xy01.w, xy23.y, xy23.w};

#pragma unroll
    for (int i = 0; i < 4; ++i) {
      const float ax = cxp[i], ay = cyp[i], az = czp[i];
      const float w  = wp[i];
      // l1 term
      const float dx = ax - gxp[i], dy = ay - gyp[i], dz = az - gzp[i];
      s1 += w * sqrtf(dx*dx + dy*dy + dz*dz);
      // uncertainty term
      const float du = w - up[i];
      su += du * du;
      // reprojection term
      const float X = r00*ax + r01*ay + r02*az + t0;
      const float Y = r10*ax + r11*ay + r12*az + t1;
      const float Z = r20*ax + r21*ay + r22*az + t2;
      const float pxi = fl * X / Z + ppx;
      const float pyi = fl * Y / Z + ppy;
      const float ex = xs[i] - pxi, ey = ys[i] - pyi;
      s2 += w * sqrtf(ex*ex + ey*ey);
    }

    // Everyone must be done reading buf[t&1] before it is refilled.
    __syncthreads();

    if (t + 2 < NTILES) {
      const int pn = p0 + (t + 2) * TILE_PTS;
      const unsigned rbase = smb + (unsigned)((t & 1) * REGION_BYTES);
      if (is_wave0) tdm_load_tile(rbase, c0 + pn, 3u);
      const unsigned lb = rbase + 3u * SLOT_BYTES + tid * 16;
      async_ld_b128(lb + 0u * SLOT_BYTES, q3 + 0 * N_SZ + pn + tid * 4);
      async_ld_b128(lb + 1u * SLOT_BYTES, q3 + 1 * N_SZ + pn + tid * 4);
      async_ld_b128(lb + 2u * SLOT_BYTES, q3 + 2 * N_SZ + pn + tid * 4);
    }
  }

  // wave32 reduction (warpSize == 32 on gfx1250), then cross-wave via LDS.
#pragma unroll
  for (int o = 16; o > 0; o >>= 1) {
    s1 += __shfl_xor(s1, o, 32);
    su += __shfl_xor(su, o, 32);
    s2 += __shfl_xor(s2, o, 32);
  }
  const int wid = tid >> 5, lane = tid & 31;
  if (lane == 0) { red[0][wid] = s1; red[1][wid] = su; red[2][wid] = s2; }
  __syncthreads();
  if (tid == 0) {
    float a = 0.f, u = 0.f, c = 0.f;
#pragma unroll
    for (int i = 0; i < 8; ++i) { a += red[0][i]; u += red[1][i]; c += red[2][i]; }
    unsafeAtomicAdd(&ws[0], a);       // sum valids * ||c - p||
    unsafeAtomicAdd(&ws[1], u);       // sum (valids - coord_pd1)^2
    unsafeAtomicAdd(&ws[2 + b], c);   // per-batch sum valids * reproj_err
  }
}

__global__ __launch_bounds__(256) void fl_criterion_finalize(
    const float* __restrict__ ws, const int* __restrict__ is_real_label,
    const int* __restrict__ epoch, float* __restrict__ out) {
  const int b = threadIdx.x; // exactly B_SZ threads
  float per = ws[2 + b] * (1.0f / (float)N_SZ);
  float m   = (is_real_label[b] != 0) ? 1.0f : 0.0f;
  float pm  = per * m;
#pragma unroll
  for (int o = 16; o > 0; o >>= 1) {
    m  += __shfl_xor(m,  o, 32);
    pm += __shfl_xor(pm, o, 32);
  }
  __shared__ float red[2][8];
  const int wid = threadIdx.x >> 5, lane = threadIdx.x & 31;
  if (lane == 0) { red[0][wid] = m; red[1][wid] = pm; }
  __syncthreads();
  if (threadIdx.x == 0) {
    float cnt = 0.f, s = 0.f;
#pragma unroll
    for (int i = 0; i < 8; ++i) { cnt += red[0][i]; s += red[1][i]; }
    const float l1  = ws[0] * (1.0f / (float)B_SZ);
    const float unc = sqrtf(ws[1]) * (1.0f / (float)B_SZ);
    const float l2  = (cnt > 0.f) ? (s / fmaxf(cnt, 1.0f)) : 0.0f;
    const float total = (epoch[0] >= START_LR) ? (l1 + unc + COEF_LR * l2)
                                               : (l1 + unc);
    out[0] = total;
    out[1] = l1;
    out[2] = l2;
    out[3] = unc;
  }
}

extern "C" void kernel_launch(void* const* d_in, const int* in_sizes, int n_in,
                              void* d_out, int out_size, void* d_ws, size_t ws_size,
                              hipStream_t stream) {
  (void)in_sizes; (void)n_in; (void)out_size; (void)ws_size;
  const float* coord_pd0 = (const float*)d_in[0];
  const float* coord_pd1 = (const float*)d_in[1];
  const float* p3Ds      = (const float*)d_in[2];
  const float* p3D_ids   = (const float*)d_in[3];
  const float* xys       = (const float*)d_in[4];
  const float* pose      = (const float*)d_in[5];
  const float* camera    = (const float*)d_in[6];
  const int*   is_real   = (const int*)d_in[7];
  const int*   epoch     = (const int*)d_in[8];
  float* ws  = (float*)d_ws;
  float* out = (float*)d_out;

  // Scratch layout: ws[0]=l1 numerator, ws[1]=unc numerator, ws[2..2+B) per-batch l2 sums.
  hipMemsetAsync(d_ws, 0, (2 + B_SZ) * sizeof(float), stream);

  fl_criterion_main<<<dim3(B_SZ * BLOCKS_PER_BATCH), dim3(256), 0, stream>>>(
      coord_pd0, coord_pd1, p3Ds, p3D_ids, xys, pose, camera, ws);

  fl_criterion_finalize<<<dim3(1), dim3(B_SZ), 0, stream>>>(ws, is_real, epoch, out);
}